// GATGraphClassifier_49289044689472
// MI455X (gfx1250) — compile-verified
//
#include <hip/hip_runtime.h>
#include <math.h>

// Problem constants (match reference)
#define NNODES   50000
#define NEDGES   1000000
#define INF_     64
#define HIDC     64
#define HEADS_   4
#define EDIM     2
#define NG       64
#define NCLS     2
#define E2TOT    (NEDGES + NNODES)
#define NEGSLOPE 0.2f

typedef __attribute__((ext_vector_type(2))) float v2f;
typedef __attribute__((ext_vector_type(8))) float v8f;

// ---------------------------------------------------------------------------
// Helpers
// ---------------------------------------------------------------------------
__device__ __forceinline__ float atomicMaxF(float* addr, float val) {
  int* ia = (int*)addr;
  int old = __float_as_int(*addr);
  while (__int_as_float(old) < val) {
    int assumed = old;
    old = atomicCAS(ia, assumed, __float_as_int(val));
    if (old == assumed) break;
  }
  return __int_as_float(old);
}

__global__ void k_fill(float* __restrict__ p, float v, long n) {
  long i = (long)blockIdx.x * blockDim.x + threadIdx.x;
  if (i < n) p[i] = v;
}

// ---------------------------------------------------------------------------
// Self-loop attr = mean of incoming edge_attr per node
// ---------------------------------------------------------------------------
__global__ void k_degsum(const int* __restrict__ ei, const float* __restrict__ ea,
                         float* __restrict__ deg, float* __restrict__ sumea) {
  int e = blockIdx.x * blockDim.x + threadIdx.x;
  if (e >= NEDGES) return;
  int d = ei[NEDGES + e];
  atomicAdd(&deg[d], 1.0f);
  atomicAdd(&sumea[2 * d + 0], ea[2 * e + 0]);
  atomicAdd(&sumea[2 * d + 1], ea[2 * e + 1]);
}

__global__ void k_mean(const float* __restrict__ deg, const float* __restrict__ sumea,
                       float* __restrict__ meanea) {
  int i = blockIdx.x * blockDim.x + threadIdx.x;
  if (i >= NNODES) return;
  float c = fmaxf(deg[i], 1.0f);
  meanea[2 * i + 0] = sumea[2 * i + 0] / c;
  meanea[2 * i + 1] = sumea[2 * i + 1] / c;
}

// ---------------------------------------------------------------------------
// fp32 WMMA GEMM: C[M,NCOLS] = A[M,K] @ B[K,NCOLS], row-major.
// One 16x16 tile per wave, V_WMMA_F32_16X16X4_F32 chained over K.
// A tile (16xK) staged in LDS, shared by all NCOLS/16 waves of the block.
// M=50000 is exactly 3125*16 -> no bounds checks, EXEC all-ones for WMMA.
// ---------------------------------------------------------------------------
template <int K, int NCOLS>
__global__ void k_gemm_wmma(const float* __restrict__ A, const float* __restrict__ B,
                            float* __restrict__ C) {
  __shared__ float As[16 * K];
  const int m0 = blockIdx.x * 16;
  for (int i = threadIdx.x; i < 16 * K; i += blockDim.x)
    As[i] = A[(size_t)(m0 + (i / K)) * K + (i % K)];
  __syncthreads();

  const int wave = threadIdx.x >> 5;      // wave = N-tile index
  const int lane = threadIdx.x & 31;
  const int n0   = wave * 16;
  const int half = lane >> 4;             // 0: K pair {0,1}, 1: K pair {2,3}
  const int l    = lane & 15;

  // A fragment (16x4 fp32): lane l<16 -> row l, K=k0+0/1 ; lanes 16-31 -> K=k0+2/3
  const float* Ap = As + l * K + half * 2;
  // B fragment (4x16 fp32): VGPR0 row k0+half*2, VGPR1 row k0+half*2+1, N = n0+l
  const float* Bp = B + (size_t)(half * 2) * NCOLS + n0 + l;

  v8f acc = {};
#pragma unroll 4
  for (int k0 = 0; k0 < K; k0 += 4) {
    v2f a, b;
    a.x = Ap[k0];
    a.y = Ap[k0 + 1];
    b.x = Bp[(size_t)k0 * NCOLS];
    b.y = Bp[(size_t)(k0 + 1) * NCOLS];
    acc = __builtin_amdgcn_wmma_f32_16x16x4_f32(false, a, false, b, (short)0, acc,
                                                false, false);
  }
  // D layout: VGPR r -> row m0+r (lanes 0-15) / m0+r+8 (lanes 16-31), col n0+l
  float* Cp = C + (size_t)(m0 + half * 8) * NCOLS + n0 + l;
#pragma unroll
  for (int r = 0; r < 8; ++r) Cp[(size_t)r * NCOLS] = acc[r];
}

// ---------------------------------------------------------------------------
// Per-node attention coefficients: a_src[n,h] = <h[n,h,:], att_s[h,:]>, same for dst
// ---------------------------------------------------------------------------
__global__ void k_attnode(const float* __restrict__ h, const float* __restrict__ as,
                          const float* __restrict__ ad, float* __restrict__ os,
                          float* __restrict__ od, int heads) {
  int idx = blockIdx.x * blockDim.x + threadIdx.x;
  if (idx >= NNODES * heads) return;
  int n  = idx / heads;
  int hh = idx - n * heads;
  const float* hp  = h + (size_t)n * heads * HIDC + hh * HIDC;
  const float* asp = as + hh * HIDC;
  const float* adp = ad + hh * HIDC;
  float ss = 0.f, dd = 0.f;
  for (int c = 0; c < HIDC; ++c) {
    float v = hp[c];
    ss += v * asp[c];
    dd += v * adp[c];
  }
  os[idx] = ss;
  od[idx] = dd;
}

// ---------------------------------------------------------------------------
// Edge-projection reduction: a_edge[e,h] = ea0*p0[h] + ea1*p1[h]
// with p_r[h] = sum_c We[r, h*64+c] * att_e[h,c].  Layer1 -> p[0..7], layer2 -> p[8..9].
// ---------------------------------------------------------------------------
__global__ void k_edgeproj(const float* __restrict__ We1, const float* __restrict__ a1e,
                           const float* __restrict__ We2, const float* __restrict__ a2e,
                           float* __restrict__ p) {
  int t = threadIdx.x;
  if (t < 8) {
    int r = t >> 2, hh = t & 3;
    const float* w = We1 + r * (HEADS_ * HIDC) + hh * HIDC;
    const float* a = a1e + hh * HIDC;
    float s = 0.f;
    for (int c = 0; c < HIDC; ++c) s += w[c] * a[c];
    p[r * 4 + hh] = s;
  } else if (t < 10) {
    int r = t - 8;
    const float* w = We2 + r * HIDC;
    float s = 0.f;
    for (int c = 0; c < HIDC; ++c) s += w[c] * a2e[c];
    p[8 + r] = s;
  }
}

// ---------------------------------------------------------------------------
// Per-edge logits + segment max.  pe[0..H) = p0, pe[H..2H) = p1.
// ---------------------------------------------------------------------------
__global__ void k_alphamax(const int* __restrict__ ei, const float* __restrict__ ea,
                           const float* __restrict__ meanea,
                           const float* __restrict__ asrc, const float* __restrict__ adst,
                           const float* __restrict__ pe, float* __restrict__ alpha,
                           float* __restrict__ amax, int H) {
  int e = blockIdx.x * blockDim.x + threadIdx.x;
  if (e >= E2TOT) return;
  int s, d;
  float e0, e1;
  if (e < NEDGES) {
    s  = ei[e];
    d  = ei[NEDGES + e];
    e0 = ea[2 * e];
    e1 = ea[2 * e + 1];
  } else {
    s = d = e - NEDGES;
    e0 = meanea[2 * s];
    e1 = meanea[2 * s + 1];
  }
  for (int hh = 0; hh < H; ++hh) {
    float al = asrc[s * H + hh] + adst[d * H + hh] + e0 * pe[hh] + e1 * pe[H + hh];
    al = al > 0.f ? al : NEGSLOPE * al;
    alpha[(size_t)e * H + hh] = al;
    atomicMaxF(&amax[d * H + hh], al);
  }
}

// exp(alpha - max) in place + segment sum
__global__ void k_expsum(const int* __restrict__ ei, const float* __restrict__ amax,
                         float* __restrict__ alpha, float* __restrict__ den, int H) {
  int e = blockIdx.x * blockDim.x + threadIdx.x;
  if (e >= E2TOT) return;
  int d = (e < NEDGES) ? ei[NEDGES + e] : e - NEDGES;
  for (int hh = 0; hh < H; ++hh) {
    float ex = __expf(alpha[(size_t)e * H + hh] - amax[d * H + hh]);
    alpha[(size_t)e * H + hh] = ex;
    atomicAdd(&den[d * H + hh], ex);
  }
}

// out[dst] += h[src] * (ex / den[dst]); one thread per (edge, channel)
__global__ void k_aggregate(const int* __restrict__ ei, const float* __restrict__ alpha,
                            const float* __restrict__ den, const float* __restrict__ h,
                            float* __restrict__ out, int H) {
  long tid = (long)blockIdx.x * blockDim.x + threadIdx.x;
  if (tid >= (long)E2TOT * HIDC) return;
  int e = (int)(tid >> 6);
  int c = (int)(tid & 63);
  int s = (e < NEDGES) ? ei[e] : e - NEDGES;
  int d = (e < NEDGES) ? ei[NEDGES + e] : e - NEDGES;
  for (int hh = 0; hh < H; ++hh) {
    float w = alpha[(size_t)e * H + hh] / den[d * H + hh];
    atomicAdd(&out[(size_t)d * H * HIDC + hh * HIDC + c],
              h[(size_t)s * H * HIDC + hh * HIDC + c] * w);
  }
}

__global__ void k_biasrelu(float* __restrict__ x, const float* __restrict__ b, int F) {
  long i = (long)blockIdx.x * blockDim.x + threadIdx.x;
  if (i >= (long)NNODES * F) return;
  int c = (int)(i % F);
  float v = x[i] + b[c];
  x[i] = v > 0.f ? v : 0.f;
}

__global__ void k_pool(const int* __restrict__ batch, const float* __restrict__ h,
                       float* __restrict__ pool, float* __restrict__ cntg) {
  long i = (long)blockIdx.x * blockDim.x + threadIdx.x;
  if (i >= (long)NNODES * HIDC) return;
  int n = (int)(i >> 6);
  int c = (int)(i & 63);
  int g = batch[n];
  atomicAdd(&pool[g * HIDC + c], h[i]);
  if (c == 0) atomicAdd(&cntg[g], 1.0f);
}

__global__ void k_classify(const float* __restrict__ pool, const float* __restrict__ cntg,
                           const float* __restrict__ Wl, const float* __restrict__ bl,
                           float* __restrict__ out) {
  int t = blockIdx.x * blockDim.x + threadIdx.x;
  if (t >= NG * NCLS) return;
  int g = t / NCLS, k = t - g * NCLS;
  float inv = 1.0f / fmaxf(cntg[g], 1.0f);
  float s = bl[k];
  for (int c = 0; c < HIDC; ++c) s += (pool[g * HIDC + c] * inv) * Wl[c * NCLS + k];
  out[t] = s;
}

// ---------------------------------------------------------------------------
// Launch
// ---------------------------------------------------------------------------
static inline dim3 gr(long n) { return dim3((unsigned)((n + 255) / 256)); }

extern "C" void kernel_launch(void* const* d_in, const int* in_sizes, int n_in,
                              void* d_out, int out_size, void* d_ws, size_t ws_size,
                              hipStream_t stream) {
  (void)in_sizes; (void)n_in; (void)out_size; (void)ws_size;
  const float* x    = (const float*)d_in[0];
  const int*   ei   = (const int*)d_in[1];
  const float* ea   = (const float*)d_in[2];
  const int*   bat  = (const int*)d_in[3];
  const float* W1   = (const float*)d_in[4];
  const float* a1s  = (const float*)d_in[5];
  const float* a1d  = (const float*)d_in[6];
  const float* We1  = (const float*)d_in[7];
  const float* a1e  = (const float*)d_in[8];
  const float* b1   = (const float*)d_in[9];
  const float* W2   = (const float*)d_in[10];
  const float* a2s  = (const float*)d_in[11];
  const float* a2d  = (const float*)d_in[12];
  const float* We2  = (const float*)d_in[13];
  const float* a2e  = (const float*)d_in[14];
  const float* b2   = (const float*)d_in[15];
  const float* Wl   = (const float*)d_in[16];
  const float* bl   = (const float*)d_in[17];
  float* out = (float*)d_out;

  float* w   = (float*)d_ws;
  size_t off = 0;
  auto alloc = [&](size_t n) { float* p = w + off; off += n; return p; };
  float* deg    = alloc(NNODES);
  float* sumea  = alloc((size_t)NNODES * EDIM);
  float* meanea = alloc((size_t)NNODES * EDIM);
  float* pvec   = alloc(16);
  float* h1     = alloc((size_t)NNODES * HEADS_ * HIDC);
  float* asrc1  = alloc((size_t)NNODES * HEADS_);
  float* adst1  = alloc((size_t)NNODES * HEADS_);
  float* alpha1 = alloc((size_t)E2TOT * HEADS_);
  float* amax1  = alloc((size_t)NNODES * HEADS_);
  float* den1   = alloc((size_t)NNODES * HEADS_);
  float* out1   = alloc((size_t)NNODES * HEADS_ * HIDC);
  float* h2     = alloc((size_t)NNODES * HIDC);
  float* asrc2  = alloc(NNODES);
  float* adst2  = alloc(NNODES);
  float* alpha2 = alloc(E2TOT);
  float* amax2  = alloc(NNODES);
  float* den2   = alloc(NNODES);
  float* out2   = alloc((size_t)NNODES * HIDC);
  float* pool   = alloc((size_t)NG * HIDC);
  float* cntg   = alloc(NG);

  // --- self-loop mean edge attr ---
  k_fill<<<gr(NNODES * 3L), 256, 0, stream>>>(deg, 0.f, NNODES * 3L); // deg+sumea contiguous
  k_degsum<<<gr(NEDGES), 256, 0, stream>>>(ei, ea, deg, sumea);
  k_mean<<<gr(NNODES), 256, 0, stream>>>(deg, sumea, meanea);

  // --- layer 1 ---
  k_gemm_wmma<INF_, HEADS_ * HIDC><<<NNODES / 16, (HEADS_ * HIDC / 16) * 32, 0, stream>>>(x, W1, h1);
  k_attnode<<<gr((long)NNODES * HEADS_), 256, 0, stream>>>(h1, a1s, a1d, asrc1, adst1, HEADS_);
  k_edgeproj<<<1, 32, 0, stream>>>(We1, a1e, We2, a2e, pvec);
  k_fill<<<gr((long)NNODES * HEADS_), 256, 0, stream>>>(amax1, -1e30f, (long)NNODES * HEADS_);
  k_fill<<<gr((long)NNODES * HEADS_), 256, 0, stream>>>(den1, 0.f, (long)NNODES * HEADS_);
  k_fill<<<gr((long)NNODES * HEADS_ * HIDC), 256, 0, stream>>>(out1, 0.f, (long)NNODES * HEADS_ * HIDC);
  k_alphamax<<<gr(E2TOT), 256, 0, stream>>>(ei, ea, meanea, asrc1, adst1, pvec, alpha1, amax1, HEADS_);
  k_expsum<<<gr(E2TOT), 256, 0, stream>>>(ei, amax1, alpha1, den1, HEADS_);
  k_aggregate<<<gr((long)E2TOT * HIDC), 256, 0, stream>>>(ei, alpha1, den1, h1, out1, HEADS_);
  k_biasrelu<<<gr((long)NNODES * HEADS_ * HIDC), 256, 0, stream>>>(out1, b1, HEADS_ * HIDC);

  // --- layer 2 ---
  k_gemm_wmma<HEADS_ * HIDC, HIDC><<<NNODES / 16, (HIDC / 16) * 32, 0, stream>>>(out1, W2, h2);
  k_attnode<<<gr(NNODES), 256, 0, stream>>>(h2, a2s, a2d, asrc2, adst2, 1);
  k_fill<<<gr(NNODES), 256, 0, stream>>>(amax2, -1e30f, NNODES);
  k_fill<<<gr(NNODES), 256, 0, stream>>>(den2, 0.f, NNODES);
  k_fill<<<gr((long)NNODES * HIDC), 256, 0, stream>>>(out2, 0.f, (long)NNODES * HIDC);
  k_alphamax<<<gr(E2TOT), 256, 0, stream>>>(ei, ea, meanea, asrc2, adst2, pvec + 8, alpha2, amax2, 1);
  k_expsum<<<gr(E2TOT), 256, 0, stream>>>(ei, amax2, alpha2, den2, 1);
  k_aggregate<<<gr((long)E2TOT * HIDC), 256, 0, stream>>>(ei, alpha2, den2, h2, out2, 1);
  k_biasrelu<<<gr((long)NNODES * HIDC), 256, 0, stream>>>(out2, b2, HIDC);

  // --- pool + classify ---
  k_fill<<<gr(NG * HIDC + NG), 256, 0, stream>>>(pool, 0.f, (long)NG * HIDC + NG); // pool+cntg contiguous
  k_pool<<<gr((long)NNODES * HIDC), 256, 0, stream>>>(bat, out2, pool, cntg);
  k_classify<<<1, 128, 0, stream>>>(pool, cntg, Wl, bl, out);
}